// GBSCM_22986664968601
// MI455X (gfx1250) — compile-verified
//
#include <hip/hip_runtime.h>
#include <hip/hip_bf16.h>
#include <math.h>

// ---------------------------------------------------------------------------
// Problem constants (from reference): n=8, inch=256, H=W=128, g=8, c2=128
// ---------------------------------------------------------------------------
constexpr int N_   = 8;
constexpr int CIN_ = 256;
constexpr int H_   = 128;
constexpr int W_   = 128;
constexpr int HWc  = H_ * W_;          // 16384
constexpr int G_   = 8;
constexpr int C2_  = 128;
constexpr float BN_EPS_ = 1e-5f;
constexpr float EPS_COS_ = 1e-8f;

typedef __attribute__((ext_vector_type(2))) float v2f;
typedef __attribute__((ext_vector_type(8))) float v8f;

// ---------------------------------------------------------------------------
// Kernel 1: bilinear upsample 64x64 -> 128x128, align_corners=True
// ---------------------------------------------------------------------------
__global__ __launch_bounds__(256) void upsample_k(const float* __restrict__ sp,
                                                  float* __restrict__ spup) {
  size_t idx = (size_t)blockIdx.x * blockDim.x + threadIdx.x;  // N*256*HW exact
  int po = (int)(idx & (HWc - 1));
  size_t nc = idx >> 14;
  int hh = po >> 7, ww = po & 127;
  const float sc = 63.0f / 127.0f;
  float fy = hh * sc, fx = ww * sc;
  int y0 = (int)fy, x0 = (int)fx;
  int y1 = min(y0 + 1, 63), x1 = min(x0 + 1, 63);
  float wy = fy - (float)y0, wx = fx - (float)x0;
  const float* b = sp + nc * 4096;
  float v00 = b[y0 * 64 + x0], v01 = b[y0 * 64 + x1];
  float v10 = b[y1 * 64 + x0], v11 = b[y1 * 64 + x1];
  spup[idx] = (v00 * (1.f - wx) + v01 * wx) * (1.f - wy)
            + (v10 * (1.f - wx) + v11 * wx) * wy;
}

// ---------------------------------------------------------------------------
// Kernel 2: 1x1 conv as fp32 WMMA GEMM.
// Y[n,oc,p] = act( scale*sum_ic W[oc,ic]X[n,ic,p] + b ), up to 3 input
// segments for channel-concat.  One wave owns (NT*16) pixels x (MT*16)
// out-channels; K stepped by 4 via v_wmma_f32_16x16x4_f32.  NT amortizes the
// A-fragment (weight) loads; MT amortizes the B-fragment (activation) loads.
// ---------------------------------------------------------------------------
template <int MT, int NT>
__global__ __launch_bounds__(256) void conv1x1_wmma_k(
    const float* __restrict__ x0, int c0,
    const float* __restrict__ x1, int c1,
    const float* __restrict__ x2, int c2s,
    const float* __restrict__ w,      // [Cout][Ctot]
    const float* __restrict__ gamma,  // bn gamma or nullptr
    const float* __restrict__ bias,   // [Cout]
    float* __restrict__ y,            // [n][Cout][HW]
    int Cout, int Ctot, int nPixGroups, int relu) {
  int wave = blockIdx.x * (blockDim.x >> 5) + (threadIdx.x >> 5);
  int lane = threadIdx.x & 31;
  const int mGroups = Cout / (16 * MT);
  if (wave >= nPixGroups * mGroups) return;    // wave-uniform: EXEC stays full
  int pg = wave / mGroups;
  int mg = wave - pg * mGroups;
  int p0 = pg * 16 * NT;        // NT*16 <= 128 divides HW: tiles share batch
  int nb = p0 >> 14;
  int po = p0 & (HWc - 1);
  int m  = lane & 15;           // A row (out channel within 16)
  int kk = (lane >> 4) << 1;    // K pair selected by lane half
  int pc = lane & 15;           // B column (pixel within tile)

  v8f acc[MT][NT];
#pragma unroll
  for (int t = 0; t < MT; ++t)
#pragma unroll
    for (int j = 0; j < NT; ++j) acc[t][j] = (v8f){0, 0, 0, 0, 0, 0, 0, 0};

  const float* xs[3] = {x0, x1, x2};
  int cs[3] = {c0, c1, c2s};
  int woff = 0;
  for (int s = 0; s < 3; ++s) {
    const float* xp = xs[s];
    int cc = cs[s];
    if (xp == nullptr || cc == 0) continue;
    const float* xb = xp + (size_t)nb * cc * HWc + po + pc;
    for (int k0 = 0; k0 < cc; k0 += 4) {
      v2f bf[NT];
#pragma unroll
      for (int j = 0; j < NT; ++j) {
        bf[j].x = xb[(size_t)(k0 + kk) * HWc + j * 16];
        bf[j].y = xb[(size_t)(k0 + kk + 1) * HWc + j * 16];
      }
#pragma unroll
      for (int t = 0; t < MT; ++t) {
        int oc = (mg * MT + t) * 16 + m;
        const float* wp = w + (size_t)oc * Ctot + woff + k0 + kk;
        v2f af;
        af.x = wp[0];
        af.y = wp[1];
#pragma unroll
        for (int j = 0; j < NT; ++j) {
          acc[t][j] = __builtin_amdgcn_wmma_f32_16x16x4_f32(
              false, af, false, bf[j], (short)0, acc[t][j], false, false);
        }
      }
    }
    woff += cc;
  }

  // Epilogue: C/D layout -> row = (lane>>4)*8 + i, col = lane&15
  int ncol = lane & 15;
  int rbase = (lane >> 4) * 8;
  const float bnmul = rsqrtf(1.0f + BN_EPS_);
#pragma unroll
  for (int t = 0; t < MT; ++t) {
    int ocb = (mg * MT + t) * 16;
#pragma unroll
    for (int i = 0; i < 8; ++i) {
      int oc = ocb + rbase + i;
      float scl = gamma ? gamma[oc] * bnmul : 1.0f;
      float bb = bias[oc];
      float* yp = y + ((size_t)nb * Cout + oc) * HWc + po + ncol;
#pragma unroll
      for (int j = 0; j < NT; ++j) {
        float v = acc[t][j][i] * scl + bb;
        if (relu) v = fmaxf(v, 0.0f);
        yp[j * 16] = v;
      }
    }
  }
}

// ---------------------------------------------------------------------------
// Kernel 3: 3x3 conv (pad=1) as fp32 WMMA GEMM with on-the-fly im2col.
// Optionally computes input as |xa - xb| elementwise (for fdist over fdif).
// ---------------------------------------------------------------------------
template <int CIN, int MT, int NT, bool ABSD>
__global__ __launch_bounds__(256) void conv3x3_wmma_k(
    const float* __restrict__ xa, const float* __restrict__ xb,
    const float* __restrict__ w,     // [Cout][CIN*9]
    const float* __restrict__ bias,  // [Cout]
    float* __restrict__ y,           // [n][Cout][HW]
    int Cout, int nPixGroups) {
  const int K = CIN * 9;
  int wave = blockIdx.x * (blockDim.x >> 5) + (threadIdx.x >> 5);
  int lane = threadIdx.x & 31;
  const int mGroups = Cout / (16 * MT);
  if (wave >= nPixGroups * mGroups) return;
  int pg = wave / mGroups;
  int mg = wave - pg * mGroups;
  int p0 = pg * 16 * NT;       // NT*16 <= 128: all tiles stay in one image row
  int nb = p0 >> 14;
  int po = p0 & (HWc - 1);
  int hh = po >> 7;
  int wbase = (po & 127) + (lane & 15);
  int m  = lane & 15;
  int kk = (lane >> 4) << 1;

  const float* xan = xa + (size_t)nb * CIN * HWc;
  const float* xbn = ABSD ? (xb + (size_t)nb * CIN * HWc) : nullptr;

  auto loadB = [&](int k, int j) -> float {
    int ic = k / 9;
    int t9 = k - ic * 9;
    int dy = t9 / 3 - 1;
    int dx = t9 - (t9 / 3) * 3 - 1;
    int h2 = hh + dy, w2 = wbase + j * 16 + dx;
    if ((unsigned)h2 >= 128u || (unsigned)w2 >= 128u) return 0.0f;  // zero pad
    size_t o = (size_t)ic * HWc + h2 * 128 + w2;
    float v = xan[o];
    if (ABSD) v = fabsf(v - xbn[o]);
    return v;
  };

  v8f acc[MT][NT];
#pragma unroll
  for (int t = 0; t < MT; ++t)
#pragma unroll
    for (int j = 0; j < NT; ++j) acc[t][j] = (v8f){0, 0, 0, 0, 0, 0, 0, 0};

  for (int k0 = 0; k0 < K; k0 += 4) {
    v2f bf[NT];
#pragma unroll
    for (int j = 0; j < NT; ++j) {
      bf[j].x = loadB(k0 + kk, j);
      bf[j].y = loadB(k0 + kk + 1, j);
    }
#pragma unroll
    for (int t = 0; t < MT; ++t) {
      int oc = (mg * MT + t) * 16 + m;
      const float* wp = w + (size_t)oc * K + k0 + kk;
      v2f af;
      af.x = wp[0];
      af.y = wp[1];
#pragma unroll
      for (int j = 0; j < NT; ++j) {
        acc[t][j] = __builtin_amdgcn_wmma_f32_16x16x4_f32(
            false, af, false, bf[j], (short)0, acc[t][j], false, false);
      }
    }
  }

  int ncol = lane & 15;
  int rbase = (lane >> 4) * 8;
#pragma unroll
  for (int t = 0; t < MT; ++t) {
    int ocb = (mg * MT + t) * 16;
#pragma unroll
    for (int i = 0; i < 8; ++i) {
      int oc = ocb + rbase + i;
      float bb = bias[oc];
      float* yp = y + ((size_t)nb * Cout + oc) * HWc + po + ncol;
#pragma unroll
      for (int j = 0; j < NT; ++j) yp[j * 16] = acc[t][j][i] + bb;
    }
  }
}

// ---------------------------------------------------------------------------
// Kernel 4: grouped cosine similarity over 16 channels per group
// ---------------------------------------------------------------------------
__global__ __launch_bounds__(256) void cossim_k(const float* __restrict__ cp1,
                                                const float* __restrict__ sp1,
                                                float* __restrict__ sim) {
  int idx = blockIdx.x * blockDim.x + threadIdx.x;  // N*G*HW exact (1048576)
  int po = idx & (HWc - 1);
  int ng = idx >> 14;
  int gi = ng & (G_ - 1);
  int n = ng >> 3;
  size_t base = ((size_t)n * C2_ + gi * 16) * HWc + po;
  const float* a = cp1 + base;
  const float* b = sp1 + base;
  float num = 0.f, na = 0.f, nb2 = 0.f;
#pragma unroll
  for (int c = 0; c < 16; ++c) {
    float av = a[(size_t)c * HWc], bv = b[(size_t)c * HWc];
    num += av * bv;
    na += av * av;
    nb2 += bv * bv;
  }
  float d = fmaxf(sqrtf(na), EPS_COS_) * fmaxf(sqrtf(nb2), EPS_COS_);
  sim[idx] = num / d;
}

// ---------------------------------------------------------------------------
// Kernel 5: dual deformable grid-sample (bilinear, align_corners, zero pad)
// ---------------------------------------------------------------------------
__device__ inline float sample_bilin(const float* __restrict__ img, float gx,
                                     float gy) {
  float ix = (gx + 1.0f) * 0.5f * 127.0f;
  float iy = (gy + 1.0f) * 0.5f * 127.0f;
  float x0f = floorf(ix), y0f = floorf(iy);
  float wx = ix - x0f, wy = iy - y0f;
  int x0 = (int)x0f, y0 = (int)y0f;
  int x1 = x0 + 1, y1 = y0 + 1;
  auto corner = [&](int xi, int yi, float wgt) -> float {
    bool ok = (xi >= 0) && (xi < 128) && (yi >= 0) && (yi < 128);
    if (!ok) return 0.0f;
    return img[yi * 128 + xi] * wgt;
  };
  return corner(x0, y0, (1.f - wx) * (1.f - wy)) +
         corner(x1, y0, wx * (1.f - wy)) +
         corner(x0, y1, (1.f - wx) * wy) +
         corner(x1, y1, wx * wy);
}

__global__ __launch_bounds__(256) void gridsample_k(
    const float* __restrict__ cp, const float* __restrict__ spup,
    const float* __restrict__ off, float* __restrict__ out) {
  size_t idx = (size_t)blockIdx.x * blockDim.x + threadIdx.x;  // N*256*HW exact
  int po = (int)(idx & (HWc - 1));
  size_t ncf = idx >> 14;
  int c = (int)(ncf & 255);
  int n = (int)(ncf >> 8);
  int gi = c >> 5;                       // 32 channels per group
  int hh = po >> 7, ww = po & 127;
  const float inv = 2.0f / 127.0f;
  float gx0 = ww * inv - 1.0f;
  float gy0 = hh * inv - 1.0f;
  const float* ob = off + (size_t)n * 32 * HWc + po;
  float olx = ob[(size_t)(2 * gi) * HWc];
  float oly = ob[(size_t)(2 * gi + 1) * HWc];
  float ohx = ob[(size_t)(16 + 2 * gi) * HWc];
  float ohy = ob[(size_t)(16 + 2 * gi + 1) * HWc];
  const float invW = 1.0f / 128.0f;      // norm = [W, H] = [128, 128]
  const float* ic = cp + ncf * HWc;
  const float* is = spup + ncf * HWc;
  float cs = sample_bilin(ic, gx0 + olx * invW, gy0 + oly * invW);
  float ss = sample_bilin(is, gx0 + ohx * invW, gy0 + ohy * invW);
  out[idx] = cs + ss;
}

// ---------------------------------------------------------------------------
// Launcher
// ---------------------------------------------------------------------------
extern "C" void kernel_launch(void* const* d_in, const int* in_sizes, int n_in,
                              void* d_out, int out_size, void* d_ws,
                              size_t ws_size, hipStream_t stream) {
  (void)in_sizes; (void)n_in; (void)out_size; (void)ws_size;
  const float* cp      = (const float*)d_in[0];
  const float* sp      = (const float*)d_in[1];
  const float* w_cp    = (const float*)d_in[2];
  const float* g_cp    = (const float*)d_in[3];
  const float* b_cp    = (const float*)d_in[4];
  const float* w_sp    = (const float*)d_in[5];
  const float* g_sp    = (const float*)d_in[6];
  const float* b_sp    = (const float*)d_in[7];
  const float* w_redim = (const float*)d_in[8];
  const float* b_redim = (const float*)d_in[9];
  const float* w_dir   = (const float*)d_in[10];
  const float* b_dir   = (const float*)d_in[11];
  const float* w_dist  = (const float*)d_in[12];
  const float* b_dist  = (const float*)d_in[13];
  const float* w_off   = (const float*)d_in[14];
  const float* b_off   = (const float*)d_in[15];
  float* out = (float*)d_out;
  float* ws = (float*)d_ws;

  // Workspace layout (floats)
  const size_t szFull = (size_t)N_ * CIN_ * HWc;   // 33 554 432
  const size_t szHalf = (size_t)N_ * C2_ * HWc;    // 16 777 216
  const size_t szSim  = (size_t)N_ * G_ * HWc;     //  1 048 576
  const size_t szOff  = (size_t)N_ * 32 * HWc;     //  4 194 304
  float* sp_up = ws;
  float* cp1   = sp_up + szFull;
  float* sp1   = cp1 + szHalf;
  float* redim = sp1 + szHalf;
  float* sim   = redim + szHalf;
  float* fdir  = sim + szSim;
  float* fdist = fdir + szOff;
  float* off   = fdist + szOff;

  const int pixGroups2 = N_ * HWc / (16 * 2);  // NT=2 -> 4096

  // 1) upsample sp -> sp_up
  upsample_k<<<(int)(szFull / 256), 256, 0, stream>>>(sp, sp_up);

  // 2) cp1 = relu(bn(conv1x1(cp)))   [MT=4, NT=2 -> 2 m-groups -> 8192 waves]
  conv1x1_wmma_k<4, 2><<<1024, 256, 0, stream>>>(
      cp, CIN_, nullptr, 0, nullptr, 0, w_cp, g_cp, b_cp, cp1, C2_, CIN_,
      pixGroups2, 1);

  // 3) sp1 = relu(bn(conv1x1(sp_up)))
  conv1x1_wmma_k<4, 2><<<1024, 256, 0, stream>>>(
      sp_up, CIN_, nullptr, 0, nullptr, 0, w_sp, g_sp, b_sp, sp1, C2_, CIN_,
      pixGroups2, 1);

  // 4) redim = conv1x1(concat(cp1, sp1)) + b_redim
  conv1x1_wmma_k<4, 2><<<1024, 256, 0, stream>>>(
      cp1, C2_, sp1, C2_, nullptr, 0, w_redim, nullptr, b_redim, redim, C2_,
      2 * C2_, pixGroups2, 0);

  // 5) grouped cosine similarity
  cossim_k<<<(int)(szSim / 256), 256, 0, stream>>>(cp1, sp1, sim);

  // 6) fdir = conv3x3(sim) + b_dir      [Cin=8, Cout=32]
  conv3x3_wmma_k<8, 2, 2, false><<<512, 256, 0, stream>>>(
      sim, nullptr, w_dir, b_dir, fdir, 32, pixGroups2);

  // 7) fdist = conv3x3(|cp1 - sp1|) + b_dist   [Cin=128, Cout=32]
  conv3x3_wmma_k<128, 2, 2, true><<<512, 256, 0, stream>>>(
      cp1, sp1, w_dist, b_dist, fdist, 32, pixGroups2);

  // 8) off = conv1x1(concat(redim, fdir, fdist)) + b_off  [Cin=192, Cout=32]
  conv1x1_wmma_k<2, 2><<<512, 256, 0, stream>>>(
      redim, C2_, fdir, 32, fdist, 32, w_off, nullptr, b_off, off, 32, 192,
      pixGroups2, 0);

  // 9) dual grid-sample -> out
  gridsample_k<<<(int)(szFull / 256), 256, 0, stream>>>(cp, sp_up, off, out);
}